// Sampler_19267223290080
// MI455X (gfx1250) — compile-verified
//
#include <hip/hip_runtime.h>
#include <hip/hip_bf16.h>
#include <cstdint>

#define TPB      256
#define VOCABN   128000
#define SPLIT    5                         // chunks per row
#define CHUNK    (VOCABN / SPLIT)          // 25600 elements per block
#define ITERS    (CHUNK / (TPB * 4))       // 25 tiles, exact
#define NBUF     4                         // async pipeline depth (3 in flight)

struct __align__(16) F4 { float x, y, z, w; };
struct __align__(16) Partial { float gv; int gi; float sv; int si; };

// LDS byte offset of a __shared__ pointer: flat addr low 32 bits == LDS offset.
__device__ __forceinline__ unsigned lds_off(const void* p) {
    return (unsigned)(size_t)p;
}

// CDNA5 async copy: global -> LDS, 16 bytes per lane, tracked by ASYNCcnt.
// GVS mode: vdst = LDS byte address, vaddr = 32-bit byte offset, saddr = base.
// NT hint: single-pass streaming data, don't pollute caches.
__device__ __forceinline__ void async_ld_b128(unsigned lds, const float* base, unsigned byte_off) {
    asm volatile("global_load_async_to_lds_b128 %0, %1, %2 th:TH_LOAD_NT"
                 :: "v"(lds), "v"(byte_off), "s"(base)
                 : "memory");
}

__device__ __forceinline__ void take_better(float& v1, int& i1, float v2, int i2) {
    // jnp.argmax semantics: larger value wins; on tie, lower index wins.
    if (v2 > v1 || (v2 == v1 && i2 < i1)) { v1 = v2; i1 = i2; }
}

__global__ __launch_bounds__(TPB) void sampler_partial_kernel(
    const float* __restrict__ logits,
    const float* __restrict__ temps,
    const float* __restrict__ noise,
    Partial* __restrict__ parts)
{
    __shared__ __align__(16) float lbuf[NBUF][TPB * 4];   // 16 KB
    __shared__ __align__(16) float nbuf[NBUF][TPB * 4];   // 16 KB
    __shared__ float r_gv[TPB]; __shared__ int r_gi[TPB];
    __shared__ float r_sv[TPB]; __shared__ int r_si[TPB];

    const int chunk = blockIdx.x;          // 0..SPLIT-1
    const int row   = blockIdx.y;          // 0..BATCH-1
    const int tid   = threadIdx.x;

    const float* lrow = logits + (size_t)row * VOCABN + (size_t)chunk * CHUNK;
    const float* nrow = noise  + (size_t)row * VOCABN + (size_t)chunk * CHUNK;
    const int    idx0 = chunk * CHUNK;     // global index of this chunk's start

    const float T    = temps[row];
    const float invT = (T == 0.0f) ? 1.0f : (1.0f / T);   // reference's safe_t

    unsigned lA[NBUF], nA[NBUF];
#pragma unroll
    for (int b = 0; b < NBUF; ++b) {
        lA[b] = lds_off(&lbuf[b][tid * 4]);
        nA[b] = lds_off(&nbuf[b][tid * 4]);
    }

    // Prologue: stage tiles 0..2 (6 outstanding async b128 loads per wave).
#pragma unroll
    for (int t = 0; t < NBUF - 1; ++t) {
        const unsigned off = (unsigned)((t * TPB + tid) * 16);
        async_ld_b128(lA[t], lrow, off);
        async_ld_b128(nA[t], nrow, off);
    }

    float gval = -__builtin_huge_valf(); int gidx = 0;   // argmax(logits)
    float sval = -__builtin_huge_valf(); int sidx = 0;   // argmax(logits/T - log(noise))

    for (int k = 0; k < ITERS; ++k) {
        const int cur = k & (NBUF - 1);

        if (k + NBUF - 1 < ITERS) {
            // Prefetch tile k+3, then wait so that only the 3 newer tiles'
            // 6 loads remain outstanding -> tile k has landed (async loads
            // complete in order within a wave).
            const int t = k + NBUF - 1;
            const int nb = t & (NBUF - 1);
            const unsigned off = (unsigned)((t * TPB + tid) * 16);
            async_ld_b128(lA[nb], lrow, off);
            async_ld_b128(nA[nb], nrow, off);
            asm volatile("s_wait_asynccnt 0x6" ::: "memory");
        } else {
            // Drain tail with exact counts: newer tiles = ITERS-1-k.
            const int remain = ITERS - 1 - k;
            if (remain >= 2)      asm volatile("s_wait_asynccnt 0x4" ::: "memory");
            else if (remain == 1) asm volatile("s_wait_asynccnt 0x2" ::: "memory");
            else                  asm volatile("s_wait_asynccnt 0x0" ::: "memory");
        }

        const F4 lv = *(const F4*)&lbuf[cur][tid * 4];
        const F4 nv = *(const F4*)&nbuf[cur][tid * 4];
        const int base = idx0 + (k * TPB + tid) * 4;

        // Ascending index order -> strict '>' keeps first occurrence in-thread.
        {
            float l = lv.x, n = nv.x;
            if (l > gval) { gval = l; gidx = base + 0; }
            float key = __builtin_fmaf(l, invT, -__logf(n));
            if (key > sval) { sval = key; sidx = base + 0; }
        }
        {
            float l = lv.y, n = nv.y;
            if (l > gval) { gval = l; gidx = base + 1; }
            float key = __builtin_fmaf(l, invT, -__logf(n));
            if (key > sval) { sval = key; sidx = base + 1; }
        }
        {
            float l = lv.z, n = nv.z;
            if (l > gval) { gval = l; gidx = base + 2; }
            float key = __builtin_fmaf(l, invT, -__logf(n));
            if (key > sval) { sval = key; sidx = base + 2; }
        }
        {
            float l = lv.w, n = nv.w;
            if (l > gval) { gval = l; gidx = base + 3; }
            float key = __builtin_fmaf(l, invT, -__logf(n));
            if (key > sval) { sval = key; sidx = base + 3; }
        }
    }

    // Block-wide argmax reduction with index tie-break (lower index wins).
    r_gv[tid] = gval; r_gi[tid] = gidx;
    r_sv[tid] = sval; r_si[tid] = sidx;
    __syncthreads();
    for (int s = TPB / 2; s > 0; s >>= 1) {
        if (tid < s) {
            take_better(r_gv[tid], r_gi[tid], r_gv[tid + s], r_gi[tid + s]);
            take_better(r_sv[tid], r_si[tid], r_sv[tid + s], r_si[tid + s]);
        }
        __syncthreads();
    }

    if (tid == 0) {
        Partial p;
        p.gv = r_gv[0]; p.gi = r_gi[0];
        p.sv = r_sv[0]; p.si = r_si[0];
        parts[row * SPLIT + chunk] = p;
    }
}

// Combine the SPLIT partials per row; one thread per row (trivial work).
__global__ __launch_bounds__(TPB) void sampler_combine_kernel(
    const Partial* __restrict__ parts,
    const float* __restrict__ temps,
    int* __restrict__ out,
    int batch)
{
    const int row = blockIdx.x * blockDim.x + threadIdx.x;
    if (row >= batch) return;

    Partial acc = parts[row * SPLIT + 0];
    // Ascending chunk order: tie-break naturally favors the lowest index.
#pragma unroll
    for (int c = 1; c < SPLIT; ++c) {
        const Partial p = parts[row * SPLIT + c];
        take_better(acc.gv, acc.gi, p.gv, p.gi);
        take_better(acc.sv, acc.si, p.sv, p.si);
    }
    out[row] = (temps[row] == 0.0f) ? acc.gi : acc.si;
}

extern "C" void kernel_launch(void* const* d_in, const int* in_sizes, int n_in,
                              void* d_out, int out_size, void* d_ws, size_t ws_size,
                              hipStream_t stream) {
    const float* logits = (const float*)d_in[0];   // (256, 128000) f32
    const float* temps  = (const float*)d_in[1];   // (256,)        f32
    const float* noise  = (const float*)d_in[2];   // (256, 128000) f32
    int* out = (int*)d_out;                        // (256,) int32 tokens

    const int batch = in_sizes[1];                 // 256 rows
    Partial* parts = (Partial*)d_ws;               // batch*SPLIT*16 B = 20 KB

    sampler_partial_kernel<<<dim3(SPLIT, batch), dim3(TPB), 0, stream>>>(
        logits, temps, noise, parts);
    sampler_combine_kernel<<<dim3((batch + TPB - 1) / TPB), dim3(TPB), 0, stream>>>(
        parts, temps, out, batch);
}